// delta_q_params_84344567759229
// MI455X (gfx1250) — compile-verified
//
#include <hip/hip_runtime.h>
#include <hip/hip_bf16.h>

typedef __attribute__((ext_vector_type(2))) float v2f;
typedef __attribute__((ext_vector_type(8))) float v8f;

#define T_DIM 16384
#define D_DIM 100
#define H_DIM 50
#define H3    150
#define XI_STRIDE 160
#define Z_DIM 64
#define ZD2   128
#define CC_STRIDE 100
#define OUT_STRIDE 128
#define LN_EPSF 1e-6f
#define KSTEPS 25            // K = 100 = 25 * 4

// ---------------------------------------------------------------------------
// Pack Wi (both dirs) into WMMA B-fragment order with zero padding:
// dst[((nt*25 + kk)*32 + lane)*2 + {0,1}] = B[k(+half*2) (+1)][nt*16 + l15]
// ---------------------------------------------------------------------------
__global__ void pack_wi_kernel(const float* __restrict__ Wi_f,
                               const float* __restrict__ Wi_b,
                               float* __restrict__ wiP_f,
                               float* __restrict__ wiP_b)
{
    const int idx = blockIdx.x * blockDim.x + threadIdx.x;
    if (idx >= 2 * 10 * KSTEPS * 32) return;
    const int dir  = idx / (10 * KSTEPS * 32);
    const int rem  = idx % (10 * KSTEPS * 32);
    const int nt   = rem / (KSTEPS * 32);
    const int rem2 = rem % (KSTEPS * 32);
    const int kk   = rem2 / 32;
    const int lane = rem2 % 32;
    const int half = lane >> 4, l15 = lane & 15;
    const int k = kk * 4 + half * 2;
    const int n = nt * 16 + l15;
    const float* Wi = dir ? Wi_b : Wi_f;
    float*       dp = dir ? wiP_b : wiP_f;
    const bool nok = (n < H3);
    dp[(size_t)rem * 2    ] = nok ? Wi[(size_t)(k    ) * H3 + n] : 0.0f;
    dp[(size_t)rem * 2 + 1] = nok ? Wi[(size_t)(k + 1) * H3 + n] : 0.0f;
}

__global__ void pack_wd_kernel(const float* __restrict__ Wd,
                               float* __restrict__ wdP)
{
    const int idx = blockIdx.x * blockDim.x + threadIdx.x;
    if (idx >= 8 * KSTEPS * 32) return;
    const int nt   = idx / (KSTEPS * 32);
    const int rem2 = idx % (KSTEPS * 32);
    const int kk   = rem2 / 32;
    const int lane = rem2 % 32;
    const int half = lane >> 4, l15 = lane & 15;
    const int k = kk * 4 + half * 2;
    const int n = nt * 16 + l15;
    wdP[(size_t)idx * 2    ] = Wd[(size_t)(k    ) * ZD2 + n];
    wdP[(size_t)idx * 2 + 1] = Wd[(size_t)(k + 1) * ZD2 + n];
}

// ---------------------------------------------------------------------------
// Kernel 1: xi = y @ Wi (fwd) and flip(y) @ Wi (bwd), WMMA f32 16x16x4.
// Branch-free inner loop: b64 A load + b64 packed-B load + wmma.
// ---------------------------------------------------------------------------
__global__ __launch_bounds__(32)
void xi_gemm_kernel(const float* __restrict__ y,
                    const float* __restrict__ wiP_f,
                    const float* __restrict__ wiP_b,
                    float* __restrict__ xi_f,
                    float* __restrict__ xi_b)
{
    const int mt  = blockIdx.x;      // 0..1023
    const int nt  = blockIdx.y;      // 0..9
    const int dir = blockIdx.z;      // 0..1
    const float* bp = (dir ? wiP_b : wiP_f) + ((size_t)nt * KSTEPS * 32 + threadIdx.x) * 2;
    float*       xi = dir ? xi_b : xi_f;

    const int lane = threadIdx.x;
    const int half = lane >> 4;
    const int l15  = lane & 15;
    const int m0 = mt * 16;
    const int n0 = nt * 16;

    const int r_log = m0 + l15;
    const int r_mem = dir ? (T_DIM - 1 - r_log) : r_log;
    const float* ap = y + (size_t)r_mem * D_DIM + half * 2;

    v8f acc = {};
    #pragma unroll
    for (int kk = 0; kk < KSTEPS; ++kk) {
        const v2f a = *(const v2f*)(ap + kk * 4);
        const v2f b = *(const v2f*)(bp + (size_t)kk * 64);
        acc = __builtin_amdgcn_wmma_f32_16x16x4_f32(
                  false, a, false, b, (short)0, acc, false, false);
    }

    #pragma unroll
    for (int r = 0; r < 8; ++r)
        xi[(size_t)(m0 + r + half * 8) * XI_STRIDE + n0 + l15] = acc[r];
}

// ---------------------------------------------------------------------------
// Kernel 2: LayerNorm-GRU scan, 1 block per direction, 192 thr = 6 waves.
// 3 barriers/step: (1) LN stats combine, (2) pre-activation exchange,
// (3) h-state publish. Wh column in 50 VGPRs; h in LDS broadcast.
// All transcendentals are branch-free __expf forms.
// ---------------------------------------------------------------------------
__global__ __launch_bounds__(192)
void gru_scan_kernel(const float* __restrict__ xi_f,
                     const float* __restrict__ xi_b,
                     const float* __restrict__ Wh_f,
                     const float* __restrict__ Wh_b,
                     const float* __restrict__ ls_f,
                     const float* __restrict__ lb_f,
                     const float* __restrict__ ls_b,
                     const float* __restrict__ lb_b,
                     float* __restrict__ cc)
{
    const int dir = blockIdx.x;
    const float* xi = dir ? xi_b : xi_f;
    const float* Wh = dir ? Wh_b : Wh_f;
    const float* ls = dir ? ls_b : ls_f;
    const float* lb = dir ? lb_b : lb_f;

    const int tid  = threadIdx.x;      // 0..191
    const int s    = tid >> 6;         // 0=r 1=z 2=n
    const int i    = tid & 63;
    const bool act = (i < H_DIM);
    const int wave = tid >> 5;         // 0..5
    const int lane = tid & 31;
    const int j    = s * H_DIM + i;

    __shared__ float hbuf[64];
    __shared__ float abuf_r[64];       // pre-activation for r gate
    __shared__ float abuf_z[64];       // pre-activation for z gate
    __shared__ float red[6][4];

    float whc[H_DIM];
    #pragma unroll
    for (int k = 0; k < H_DIM; ++k)
        whc[k] = act ? Wh[(size_t)k * H3 + j] : 0.0f;

    float lsx = 0.f, lbx = 0.f, lsh = 0.f, lbh = 0.f;
    if (act) {
        lsx = ls[(2 * s    ) * H_DIM + i];  lbx = lb[(2 * s    ) * H_DIM + i];
        lsh = ls[(2 * s + 1) * H_DIM + i];  lbh = lb[(2 * s + 1) * H_DIM + i];
    }

    if (tid < 64) hbuf[tid] = 0.0f;
    __syncthreads();

    const float inv50 = 1.0f / 50.0f;

    for (int t = 0; t < T_DIM; ++t) {
        const float xt = act ? xi[(size_t)t * XI_STRIDE + j] : 0.0f;
        if (t + 1 < T_DIM)
            __builtin_prefetch(&xi[(size_t)(t + 1) * XI_STRIDE + j], 0, 0);

        float hh = 0.0f;
        #pragma unroll
        for (int k = 0; k < H_DIM; ++k)
            hh = fmaf(hbuf[k], whc[k], hh);

        float s0 = xt, s1 = xt * xt, s2 = hh, s3 = hh * hh;
        #pragma unroll
        for (int off = 16; off >= 1; off >>= 1) {
            s0 += __shfl_xor(s0, off, 32);
            s1 += __shfl_xor(s1, off, 32);
            s2 += __shfl_xor(s2, off, 32);
            s3 += __shfl_xor(s3, off, 32);
        }
        if (lane == 0) {
            red[wave][0] = s0; red[wave][1] = s1;
            red[wave][2] = s2; red[wave][3] = s3;
        }
        __syncthreads();                                   // (1)
        const int w0 = s * 2;
        const float sumx  = red[w0][0] + red[w0 + 1][0];
        const float sumx2 = red[w0][1] + red[w0 + 1][1];
        const float sumh  = red[w0][2] + red[w0 + 1][2];
        const float sumh2 = red[w0][3] + red[w0 + 1][3];

        const float mx = sumx * inv50;
        const float vx = fmaf(-mx, mx, sumx2 * inv50);
        const float mh = sumh * inv50;
        const float vh = fmaf(-mh, mh, sumh2 * inv50);

        const float lnx = (xt - mx) * __frsqrt_rn(vx + LN_EPSF) * lsx + lbx;
        const float lnh = (hh - mh) * __frsqrt_rn(vh + LN_EPSF) * lsh + lbh;

        if (s == 0) { if (act) abuf_r[i] = lnx + lnh; }
        else if (s == 1) { if (act) abuf_z[i] = lnx + lnh; }
        __syncthreads();                                   // (2)
        if (s == 2 && act) {
            const float r  = 1.0f / (1.0f + __expf(-abuf_r[i]));
            const float z  = 1.0f / (1.0f + __expf(-abuf_z[i]));
            const float na = lnx + r * lnh;
            // tanh(x) = 1 - 2/(e^{2x}+1): branch-free, saturates at +/-1
            const float n  = 1.0f - 2.0f / (__expf(2.0f * na) + 1.0f);
            const float hold = hbuf[i];
            const float hn = fmaf(z, hold - n, n);         // (1-z)n + z h
            hbuf[i] = hn;
            const size_t row = dir ? (size_t)(T_DIM - 1 - t) : (size_t)t;
            cc[row * CC_STRIDE + dir * H_DIM + i] = hn;
        }
        __syncthreads();                                   // (3)
    }
}

// ---------------------------------------------------------------------------
// Kernel 3: out = cc @ Wd + bd, WMMA f32 16x16x4, packed B.
// ---------------------------------------------------------------------------
__global__ __launch_bounds__(32)
void out_gemm_kernel(const float* __restrict__ cc,
                     const float* __restrict__ wdP,
                     const float* __restrict__ bd,
                     float* __restrict__ outb)
{
    const int mt = blockIdx.x;   // 0..1023
    const int nt = blockIdx.y;   // 0..7
    const int lane = threadIdx.x;
    const int half = lane >> 4;
    const int l15  = lane & 15;
    const int m0 = mt * 16, n0 = nt * 16;
    const int ncol = n0 + l15;

    const float* ap = cc + (size_t)(m0 + l15) * CC_STRIDE + half * 2;
    const float* bp = wdP + ((size_t)nt * KSTEPS * 32 + lane) * 2;

    const float bias = bd[ncol];
    v8f acc;
    #pragma unroll
    for (int r = 0; r < 8; ++r) acc[r] = bias;

    #pragma unroll
    for (int kk = 0; kk < KSTEPS; ++kk) {
        const v2f a = *(const v2f*)(ap + kk * 4);
        const v2f b = *(const v2f*)(bp + (size_t)kk * 64);
        acc = __builtin_amdgcn_wmma_f32_16x16x4_f32(
                  false, a, false, b, (short)0, acc, false, false);
    }

    #pragma unroll
    for (int r = 0; r < 8; ++r)
        outb[(size_t)(m0 + r + half * 8) * OUT_STRIDE + ncol] = acc[r];
}

// ---------------------------------------------------------------------------
// Kernel 4a: Sigma and J diagonal fills, float4 (b128) stores. HBM-bound.
// ---------------------------------------------------------------------------
__global__ void finalize_mats_kernel(const float* __restrict__ outb,
                                     float4* __restrict__ Sigma4,
                                     float4* __restrict__ J4)
{
    const size_t N4 = (size_t)T_DIM * Z_DIM * Z_DIM / 4;
    size_t idx = (size_t)blockIdx.x * blockDim.x + threadIdx.x;
    const size_t stride = (size_t)gridDim.x * blockDim.x;
    for (; idx < N4; idx += stride) {
        const size_t e = idx * 4;
        const size_t t = e >> 12;            // Z*Z = 4096
        const int rem = (int)(e & 4095);
        const int a  = rem >> 6;
        const int b0 = rem & 63;             // multiple of 4
        float4 sv = make_float4(0.f, 0.f, 0.f, 0.f);
        float4 jv = make_float4(0.f, 0.f, 0.f, 0.f);
        if (a >= b0 && a < b0 + 4) {
            const float ev = expf(outb[t * OUT_STRIDE + Z_DIM + a]);
            const int p = a - b0;
            ((float*)&sv)[p] = ev + 1e-6f;
            ((float*)&jv)[p] = 1.0f / ev;
        }
        Sigma4[idx] = sv;
        J4[idx]     = jv;
    }
}

// ---------------------------------------------------------------------------
// Kernel 4b: mu and h = mu / exp(v).
// ---------------------------------------------------------------------------
__global__ void finalize_vecs_kernel(const float* __restrict__ outb,
                                     float* __restrict__ mu,
                                     float* __restrict__ hout)
{
    const size_t N = (size_t)T_DIM * Z_DIM;
    const size_t idx = (size_t)blockIdx.x * blockDim.x + threadIdx.x;
    if (idx >= N) return;
    const size_t t = idx >> 6;
    const int q = (int)(idx & 63);
    const float m  = outb[t * OUT_STRIDE + q];
    const float ev = expf(outb[t * OUT_STRIDE + Z_DIM + q]);
    mu[idx]   = m;
    hout[idx] = m / ev;
}

// ---------------------------------------------------------------------------
extern "C" void kernel_launch(void* const* d_in, const int* in_sizes, int n_in,
                              void* d_out, int out_size, void* d_ws, size_t ws_size,
                              hipStream_t stream)
{
    (void)in_sizes; (void)n_in; (void)out_size; (void)ws_size;

    const float* y    = (const float*)d_in[0];
    const float* Wi_f = (const float*)d_in[1];
    const float* Wh_f = (const float*)d_in[2];
    const float* ls_f = (const float*)d_in[3];
    const float* lb_f = (const float*)d_in[4];
    const float* Wi_b = (const float*)d_in[5];
    const float* Wh_b = (const float*)d_in[6];
    const float* ls_b = (const float*)d_in[7];
    const float* lb_b = (const float*)d_in[8];
    const float* Wd   = (const float*)d_in[9];
    const float* bd   = (const float*)d_in[10];

    // Workspace layout (floats):
    float* ws    = (float*)d_ws;
    float* xi_f  = ws;                                    // T*160
    float* xi_b  = xi_f  + (size_t)T_DIM * XI_STRIDE;     // T*160
    float* cc    = xi_b  + (size_t)T_DIM * XI_STRIDE;     // T*100
    float* outb  = cc    + (size_t)T_DIM * CC_STRIDE;     // T*128
    float* wiP_f = outb  + (size_t)T_DIM * OUT_STRIDE;    // 10*25*32*2
    float* wiP_b = wiP_f + (size_t)10 * KSTEPS * 32 * 2;
    float* wdP   = wiP_b + (size_t)10 * KSTEPS * 32 * 2;  // 8*25*32*2

    // Output layout (flat concat, return order): Sigma, mu, J, h
    float* Sigma = (float*)d_out;
    float* mu    = Sigma + (size_t)T_DIM * Z_DIM * Z_DIM;
    float* J     = mu    + (size_t)T_DIM * Z_DIM;
    float* hout  = J     + (size_t)T_DIM * Z_DIM * Z_DIM;

    pack_wi_kernel<<<(2 * 10 * KSTEPS * 32 + 255) / 256, 256, 0, stream>>>(
        Wi_f, Wi_b, wiP_f, wiP_b);
    pack_wd_kernel<<<(8 * KSTEPS * 32 + 255) / 256, 256, 0, stream>>>(Wd, wdP);

    dim3 g1(T_DIM / 16, 10, 2);
    xi_gemm_kernel<<<g1, 32, 0, stream>>>(y, wiP_f, wiP_b, xi_f, xi_b);

    gru_scan_kernel<<<2, 192, 0, stream>>>(xi_f, xi_b, Wh_f, Wh_b,
                                           ls_f, lb_f, ls_b, lb_b, cc);

    dim3 g3(T_DIM / 16, 8);
    out_gemm_kernel<<<g3, 32, 0, stream>>>(cc, wdP, bd, outb);

    finalize_mats_kernel<<<8192, 256, 0, stream>>>(
        outb, (float4*)Sigma, (float4*)J);
    finalize_vecs_kernel<<<(T_DIM * Z_DIM) / 256, 256, 0, stream>>>(outb, mu, hout);
}